// RGCNEncoder_2430951490178
// MI455X (gfx1250) — compile-verified
//
#include <hip/hip_runtime.h>

#define R_REL   8
#define D_DIM   64
#define G_GRAPH 64

typedef __attribute__((ext_vector_type(2))) float v2f;
typedef __attribute__((ext_vector_type(8))) float v8f;

// D = A(16x4 f32) * B(4x16 f32) + C  — fp32 matrix core
__device__ __forceinline__ v8f wmma_f32_16x16x4(v2f a, v2f b, v8f c) {
  return __builtin_amdgcn_wmma_f32_16x16x4_f32(
      /*neg_a=*/false, a, /*neg_b=*/false, b,
      /*c_mod=*/(short)0, c, /*reuse_a=*/false, /*reuse_b=*/false);
}

// ---------------------------------------------------------------- utilities
__global__ void zero_u32(unsigned int* p, int n) {
  int i = blockIdx.x * blockDim.x + threadIdx.x;
  if (i < n) p[i] = 0u;
}

// per-(dst,relation) in-degree counts (shared by both layers)
__global__ void count_edges(const int* __restrict__ dst, const int* __restrict__ et,
                            int* __restrict__ cnt, int E) {
  int e = blockIdx.x * blockDim.x + threadIdx.x;
  if (e < E) atomicAdd(&cnt[dst[e] * R_REL + et[e]], 1);
}

// ------------------------------------------------- per-relation transform GEMM
// Y[n, r*64+h] = sum_k X[n,k] * W[r][k][h]     (optionally relu(X) for layer 2)
// block = 256 threads = 8 waves; grid = (N/16 tiles, 4); wave -> one 16x16 tile
template <bool RELU>
__global__ void gemm_rel(const float* __restrict__ X, const float* __restrict__ W,
                         float* __restrict__ Y, int N) {
  __shared__ float As[16 * 65];                 // 16x64 A tile, stride 65 (no bank conflicts)
  const int tid   = threadIdx.x;
  const int nbase = blockIdx.x * 16;

  for (int i = tid; i < 16 * 64; i += 256) {    // stage A once per block (8 waves reuse)
    int row = i >> 6, k = i & 63;
    int n = nbase + row; if (n >= N) n = N - 1;
    float v = X[(size_t)n * 64 + k];
    if (RELU) v = v > 0.f ? v : 0.f;
    As[row * 65 + k] = v;
  }
  __syncthreads();

  const int wave = tid >> 5, lane = tid & 31;
  const int half = lane >> 4, l16 = lane & 15;
  const int ct    = blockIdx.y * 8 + wave;      // column tile 0..31 over R*64 = 512 cols
  const int r     = ct >> 2;
  const int hbase = (ct & 3) * 16;
  const float* Wr = W + (size_t)r * 64 * 64;    // [64 x 64], k-major

  v8f acc = {};
#pragma unroll
  for (int k0 = 0; k0 < 64; k0 += 4) {
    const int kA = k0 + 2 * half;               // ISA layout: K = k0 + 2*(lane/16) + i
    v2f a, b;
    a.x = As[l16 * 65 + kA];
    a.y = As[l16 * 65 + kA + 1];
    b.x = Wr[(size_t)kA * 64 + hbase + l16];
    b.y = Wr[(size_t)(kA + 1) * 64 + hbase + l16];
    acc = wmma_f32_16x16x4(a, b, acc);
  }
#pragma unroll
  for (int v = 0; v < 8; ++v) {                 // D: row = v + 8*(lane/16), col = lane%16
    int row = nbase + v + 8 * half;
    if (row < N) Y[(size_t)row * (R_REL * 64) + ct * 16 + l16] = acc[v];
  }
}

// ------------------------------------------------- root GEMM (+bias) -> Out init
// Out[n,h] = (relu?)X[n,:] @ Wroot[:,h] + bias[h]
// block = 128 threads = 4 waves; grid = N/16 tiles
template <bool RELU>
__global__ void gemm_root(const float* __restrict__ X, const float* __restrict__ Wroot,
                          const float* __restrict__ bias, float* __restrict__ Out, int N) {
  __shared__ float As[16 * 65];
  const int tid   = threadIdx.x;
  const int nbase = blockIdx.x * 16;

  for (int i = tid; i < 16 * 64; i += 128) {
    int row = i >> 6, k = i & 63;
    int n = nbase + row; if (n >= N) n = N - 1;
    float v = X[(size_t)n * 64 + k];
    if (RELU) v = v > 0.f ? v : 0.f;
    As[row * 65 + k] = v;
  }
  __syncthreads();

  const int wave = tid >> 5, lane = tid & 31;
  const int half = lane >> 4, l16 = lane & 15;
  const int hbase = wave * 16;

  v8f acc = {};
#pragma unroll
  for (int k0 = 0; k0 < 64; k0 += 4) {
    const int kA = k0 + 2 * half;
    v2f a, b;
    a.x = As[l16 * 65 + kA];
    a.y = As[l16 * 65 + kA + 1];
    b.x = Wroot[(size_t)kA * 64 + hbase + l16];
    b.y = Wroot[(size_t)(kA + 1) * 64 + hbase + l16];
    acc = wmma_f32_16x16x4(a, b, acc);
  }
  const float bb = bias[hbase + l16];
#pragma unroll
  for (int v = 0; v < 8; ++v) {
    int row = nbase + v + 8 * half;
    if (row < N) Out[(size_t)row * 64 + hbase + l16] = acc[v] + bb;
  }
}

// ------------------------------------------------- edge scatter (mean-weighted)
// Out[dst] += y[src, etype, :] / cnt[dst, etype]   — one wave per edge,
// 25.6 MB accumulator stays L2-resident on the 192 MB L2.
__global__ void scatter_edges(const float* __restrict__ Y, const int* __restrict__ src,
                              const int* __restrict__ dst, const int* __restrict__ et,
                              const int* __restrict__ cnt, float* __restrict__ Out, int E) {
  const int gid  = blockIdx.x * blockDim.x + threadIdx.x;
  const int e    = gid >> 5;
  const int lane = gid & 31;
  if (e >= E) return;
  const int s = src[e], d = dst[e], r = et[e];
  const int c = cnt[d * R_REL + r];
  const float inv = 1.0f / (float)(c > 0 ? c : 1);
  const float* yrow = Y + (size_t)s * (R_REL * 64) + r * 64;
  atomicAdd(&Out[(size_t)d * 64 + lane],      yrow[lane]      * inv);
  atomicAdd(&Out[(size_t)d * 64 + lane + 32], yrow[lane + 32] * inv);
}

// ------------------------------------------------- global mean pool
__global__ void pool_sum(const float* __restrict__ H, const int* __restrict__ batch,
                         float* __restrict__ gsum, float* __restrict__ gcnt, int N) {
  const int gid  = blockIdx.x * blockDim.x + threadIdx.x;
  const int n    = gid >> 5;
  const int lane = gid & 31;
  if (n >= N) return;
  const int g = batch[n];
  atomicAdd(&gsum[g * 64 + lane],      H[(size_t)n * 64 + lane]);
  atomicAdd(&gsum[g * 64 + lane + 32], H[(size_t)n * 64 + lane + 32]);
  if (lane == 0) atomicAdd(&gcnt[g], 1.0f);
}

__global__ void pool_finalize(const float* __restrict__ gsum, const float* __restrict__ gcnt,
                              float* __restrict__ out) {
  int i = blockIdx.x * blockDim.x + threadIdx.x;
  if (i < G_GRAPH * 64) {
    float c = gcnt[i >> 6];
    out[i] = gsum[i] / (c > 0.f ? c : 1.f);
  }
}

// ---------------------------------------------------------------- launch
extern "C" void kernel_launch(void* const* d_in, const int* in_sizes, int n_in,
                              void* d_out, int out_size, void* d_ws, size_t ws_size,
                              hipStream_t stream) {
  const float* x     = (const float*)d_in[0];
  const int*   eidx  = (const int*)d_in[1];   // [2, E]
  const int*   etyp  = (const int*)d_in[2];   // [E]
  const int*   batch = (const int*)d_in[3];   // [N]
  const float* W1    = (const float*)d_in[4];
  const float* root1 = (const float*)d_in[5];
  const float* b1    = (const float*)d_in[6];
  const float* W2    = (const float*)d_in[7];
  const float* root2 = (const float*)d_in[8];
  const float* b2    = (const float*)d_in[9];
  float*       out   = (float*)d_out;

  const int N = in_sizes[0] / D_DIM;
  const int E = in_sizes[2];
  const int* src  = eidx;
  const int* dstp = eidx + E;

  // workspace carve-up (256B aligned): y | h1 | h2 | cnt | gsum | gcnt
  char* ws = (char*)d_ws;
  size_t off = 0;
  auto carve = [&](size_t bytes) -> void* {
    void* p = ws + off;
    off = (off + bytes + 255) & ~(size_t)255;
    return p;
  };
  float* y    = (float*)carve((size_t)N * R_REL * 64 * sizeof(float)); // 204.8 MB
  float* h1   = (float*)carve((size_t)N * 64 * sizeof(float));         // 25.6 MB
  float* h2   = (float*)carve((size_t)N * 64 * sizeof(float));         // 25.6 MB
  int*   cnt  = (int*)carve((size_t)N * R_REL * sizeof(int));          // 3.2 MB
  float* gsum = (float*)carve((size_t)G_GRAPH * 64 * sizeof(float));
  float* gcnt = (float*)carve((size_t)G_GRAPH * sizeof(float));
  (void)ws_size; (void)n_in; (void)out_size;

  const int ntiles = (N + 15) / 16;

  // init accumulators (counts + pool; h1/h2/y are fully overwritten before use)
  zero_u32<<<(N * R_REL + 255) / 256, 256, 0, stream>>>((unsigned int*)cnt, N * R_REL);
  zero_u32<<<(G_GRAPH * 65 + 255) / 256, 256, 0, stream>>>((unsigned int*)gsum,
                                                           G_GRAPH * 64 + G_GRAPH);
  count_edges<<<(E + 255) / 256, 256, 0, stream>>>(dstp, etyp, cnt, E);

  // ---- layer 1 ----
  gemm_rel<false><<<dim3(ntiles, 4), 256, 0, stream>>>(x, W1, y, N);
  gemm_root<false><<<ntiles, 128, 0, stream>>>(x, root1, b1, h1, N);
  scatter_edges<<<(E * 32 + 255) / 256, 256, 0, stream>>>(y, src, dstp, etyp, cnt, h1, E);

  // ---- layer 2 (ReLU folded into A-tile staging loads) ----
  gemm_rel<true><<<dim3(ntiles, 4), 256, 0, stream>>>(h1, W2, y, N);
  gemm_root<true><<<ntiles, 128, 0, stream>>>(h1, root2, b2, h2, N);
  scatter_edges<<<(E * 32 + 255) / 256, 256, 0, stream>>>(y, src, dstp, etyp, cnt, h2, E);

  // ---- global mean pool ----
  pool_sum<<<(N * 32 + 255) / 256, 256, 0, stream>>>(h2, batch, gsum, gcnt, N);
  pool_finalize<<<(G_GRAPH * 64 + 255) / 256, 256, 0, stream>>>(gsum, gcnt, out);
}